// KDTreeRBF_26345329393933
// MI455X (gfx1250) — compile-verified
//
#include <hip/hip_runtime.h>
#include <stdint.h>

// KD-tree RBF scatter-gather interpolation, MI455X (gfx1250, wave32).
//
// Roofline: idx stream = 256 MB (dominant), points+out = 24 MB -> ~12 us floor
// at 23.3 TB/s. Compute is ~67M (i,k) pairs; WMMA is inapplicable (the gathered
// ctrl/disp rows are data-dependent per output row -> no shared B operand), so
// the win is LDS-resident tables + minimal VALU per pair:
//
//   w = exp(-|p-c|^2 / (2s^2)) = exp2( C|p|^2 + C|c|^2 + p . (-2C c) ),
//   C = -log2(e)/(2 s^2)
//
// Table row: t.xyz = -2C*c, t.w = C*|c|^2  (sentinel row: (0,0,0,-1e30) -> w=0)
// Per pair: 1 shift + 1 add + 3 fma + v_exp_f32 + 4 fma/add + 2x ds_load_b128.

#define SIGMA      10.0f
#define C_LOG2E    (-0.72134752044f / (SIGMA * SIGMA))  // -log2(e)/(2*sigma^2)
#define KNEIGH     64
#define MAX_CTRL   511   // table has MAX_CTRL+1 float4 slots (sentinel incl.)

#define HAS_ASYNC_LDS  __has_builtin(__builtin_amdgcn_global_load_async_to_lds_b32)
#define HAS_WAIT_ASYNC __has_builtin(__builtin_amdgcn_s_wait_asynccnt)

typedef __attribute__((address_space(1))) int gmem_i32;   // global
typedef __attribute__((address_space(3))) int lds_i32;    // LDS

__device__ __forceinline__ void wait_asynccnt0() {
#if HAS_WAIT_ASYNC
    __builtin_amdgcn_s_wait_asynccnt(0);
#else
    asm volatile("s_wait_asynccnt 0" ::: "memory");
#endif
}

__device__ __forceinline__ float fast_exp2(float x) {
#if __has_builtin(__builtin_amdgcn_exp2f)
    return __builtin_amdgcn_exp2f(x);   // v_exp_f32
#else
    return exp2f(x);
#endif
}

__global__ __launch_bounds__(256)
void rbf_gather_kernel(const float* __restrict__ points,
                       const float* __restrict__ ctrl,
                       const float* __restrict__ disp,
                       const int*   __restrict__ idx,
                       float* __restrict__ out,
                       int n, int n_ctrl)
{
    __shared__ float4 s_ctrl[MAX_CTRL + 1];   // 8 KB, transformed rows
    __shared__ float4 s_disp[MAX_CTRL + 1];   // 8 KB, raw displacement rows

    const int i = blockIdx.x * blockDim.x + threadIdx.x;

    // Kick the idx stream early (speculative prefetch -> global_prefetch_b8);
    // overlaps the DRAM fetch with the LDS table fill below.
    __builtin_prefetch(idx + (size_t)i * KNEIGH, 0, 3);

    // ---- Fill LDS tables -------------------------------------------------
    // Sentinel rows (index == n_ctrl): ctrl row forces arg = -1e30 -> w = 0
    // (exact reference mask semantics); disp row zero.
    if (threadIdx.x == 0) {
        s_ctrl[n_ctrl] = make_float4(0.f, 0.f, 0.f, -1e30f);
        s_disp[n_ctrl] = make_float4(0.f, 0.f, 0.f, 0.f);
    }

#if HAS_ASYNC_LDS
    // disp table: gfx1250 async global->LDS B32 copies on ASYNCcnt
    // (no VGPR round trip; .w lanes are never read).
    for (int e = threadIdx.x; e < 3 * n_ctrl; e += blockDim.x) {
        const int c = e / 3;
        const int f = e - 3 * c;
        gmem_i32* gsrc = (gmem_i32*)(uintptr_t)(disp + e);
        lds_i32*  ldst = (lds_i32*)(uint32_t)(uintptr_t)((float*)&s_disp[c] + f);
        __builtin_amdgcn_global_load_async_to_lds_b32(gsrc, ldst, 0, 0);
    }
#else
    for (int c = threadIdx.x; c < n_ctrl; c += blockDim.x)
        s_disp[c] = make_float4(disp[3*c+0], disp[3*c+1], disp[3*c+2], 0.f);
#endif

    // ctrl table: load raw xyz, store the transformed quadratic-form row.
    for (int c = threadIdx.x; c < n_ctrl; c += blockDim.x) {
        const float cx = ctrl[3*c+0], cy = ctrl[3*c+1], cz = ctrl[3*c+2];
        const float n2 = fmaf(cx, cx, fmaf(cy, cy, cz * cz));
        const float k  = -2.0f * C_LOG2E;
        s_ctrl[c] = make_float4(k * cx, k * cy, k * cz, C_LOG2E * n2);
    }

#if HAS_ASYNC_LDS
    wait_asynccnt0();
#endif
    __syncthreads();

    if (i >= n) return;

    const float px = points[3*(size_t)i + 0];
    const float py = points[3*(size_t)i + 1];
    const float pz = points[3*(size_t)i + 2];
    const float a  = C_LOG2E * fmaf(px, px, fmaf(py, py, pz * pz)); // C*|p|^2

    // idx row: 256 B, 16-byte aligned -> stream as 16x int4 (global_load_b128)
    const int4* idx4 = reinterpret_cast<const int4*>(idx + (size_t)i * KNEIGH);

    float wsum = 0.f;
    float mx = 0.f, my = 0.f, mz = 0.f;

    #pragma unroll 4
    for (int c = 0; c < KNEIGH / 4; ++c) {
        const int4 j4 = idx4[c];
        const int js[4] = { j4.x, j4.y, j4.z, j4.w };
        #pragma unroll
        for (int u = 0; u < 4; ++u) {
            const int j = js[u];
            const float4 t = s_ctrl[j];               // ds_load_b128
            const float4 d = s_disp[j];               // ds_load_b128 (+8192)
            float arg = a + t.w;                      // C(|p|^2 + |c|^2)
            arg = fmaf(px, t.x, arg);
            arg = fmaf(py, t.y, arg);
            arg = fmaf(pz, t.z, arg);                 // + p.(-2C c) = C r2
            const float w = fast_exp2(arg);           // v_exp_f32; sentinel->0
            wsum += w;
            mx = fmaf(w, d.x, mx);
            my = fmaf(w, d.y, my);
            mz = fmaf(w, d.z, mz);
        }
    }

    // reference: w_sum = (w_sum < 1e-5) ? 1 : w_sum; motion = sum(w*cd)/w_sum
    const float inv = (wsum < 1e-5f) ? 1.0f : __frcp_rn(wsum);
    out[3*(size_t)i + 0] = mx * inv;
    out[3*(size_t)i + 1] = my * inv;
    out[3*(size_t)i + 2] = mz * inv;
}

extern "C" void kernel_launch(void* const* d_in, const int* in_sizes, int n_in,
                              void* d_out, int out_size, void* d_ws, size_t ws_size,
                              hipStream_t stream)
{
    const float* points = (const float*)d_in[0];
    const float* ctrl   = (const float*)d_in[1];
    const float* disp   = (const float*)d_in[2];
    const int*   idx    = (const int*)d_in[3];
    float*       out    = (float*)d_out;

    const int n      = in_sizes[0] / 3;   // 1048576
    const int n_ctrl = in_sizes[1] / 3;   // 500 (must be <= MAX_CTRL)

    const int block = 256;                // 8 wave32 waves per block
    const int grid  = (n + block - 1) / block;

    hipLaunchKernelGGL(rbf_gather_kernel, dim3(grid), dim3(block), 0, stream,
                       points, ctrl, disp, idx, out, n, n_ctrl);
}